// PositionalGATLayer_16123307229579
// MI455X (gfx1250) — compile-verified
//
#include <hip/hip_runtime.h>

// Problem constants (fixed by the reference)
#define NN   50000
#define EE   800000
#define HH   4
#define DD   64
#define PP   16
#define CIN  256      // input feature width
#define CC   240      // content dims (CIN - PP)
#define HD   256      // H*D output width

#define KCHUNK   16
#define A_STRIDE 20   // words; 80B rows -> 16B-aligned float4 stores, conflict-free b64 reads
#define B_STRIDE 144  // words; 144 mod 64 = 16 -> lane halves hit disjoint bank groups

typedef float v2f __attribute__((ext_vector_type(2)));
typedef float v8f __attribute__((ext_vector_type(8)));

// ---------- order-preserving float <-> uint encoding for atomicMax ----------
__device__ __forceinline__ unsigned enc_f32(float x) {
    unsigned u = __float_as_uint(x);
    return (u & 0x80000000u) ? ~u : (u | 0x80000000u);
}
__device__ __forceinline__ float dec_f32(unsigned k) {
    unsigned u = (k & 0x80000000u) ? (k & 0x7FFFFFFFu) : ~k;
    return __uint_as_float(u);
}
#define ENC_NEG_INF 0x007FFFFFu   // enc_f32(-inf)

// ---------- init: d_out = feat (residual seed for scatter atomics) ----------
__global__ void init_out_kernel(const float4* __restrict__ feat, float4* __restrict__ out) {
    int i = blockIdx.x * blockDim.x + threadIdx.x;
    if (i < NN * CIN / 4) out[i] = feat[i];
}

__global__ void init_nh_kernel(unsigned* __restrict__ emaxk, float* __restrict__ denom) {
    int i = blockIdx.x * blockDim.x + threadIdx.x;
    if (i < NN * HH) { emaxk[i] = ENC_NEG_INF; denom[i] = 0.0f; }
}

// ---------- GEMM: ft = content @ W via V_WMMA_F32_16X16X4_F32 ----------
// Block: one 16-row m-tile x 128 cols; 8 waves, one 16x16 n-tile each.
// Double-buffered LDS k-chunk pipeline (global->regs overlaps WMMA, regs->LDS, 1 barrier/chunk).
__global__ void __launch_bounds__(256) gemm_wmma_kernel(const float* __restrict__ feat,
                                                        const float* __restrict__ W,
                                                        float* __restrict__ ft) {
    __shared__ float Abuf[2][16 * A_STRIDE];   // 2.5 KB
    __shared__ float Bbuf[2][KCHUNK * B_STRIDE]; // 18 KB

    const int tid  = threadIdx.x;
    const int wave = tid >> 5;
    const int lane = tid & 31;
    const int half = lane >> 4;                // selects K pair {0,1} vs {2,3}
    const int r    = lane & 15;                // M (A frag) / N (B frag)

    const int m0  = (blockIdx.x >> 1) * 16;    // 3125 m-tiles
    const int nb0 = (blockIdx.x & 1) * 128;    // two 128-col halves
    const int n0  = nb0 + wave * 16;

    // cooperative-loader coordinates
    const int brow0 = tid >> 5;                // B: j = tid      -> row, col4
    const int bcol0 = (tid & 31) * 4;
    const int brow1 = (tid + 256) >> 5;        // B: j = tid+256
    const int bcol1 = (tid & 31) * 4;
    const int arow  = tid >> 2;                // A: threads 0..63
    const int acol  = (tid & 3) * 4;

    // ---- prologue: stage chunk 0 into buffer 0 ----
    float4 bf0 = *(const float4*)(W + (size_t)brow0 * HD + nb0 + bcol0);
    float4 bf1 = *(const float4*)(W + (size_t)brow1 * HD + nb0 + bcol1);
    float4 af;
    if (tid < 64) af = *(const float4*)(feat + (size_t)(m0 + arow) * CIN + acol);
    *(float4*)(&Bbuf[0][brow0 * B_STRIDE + bcol0]) = bf0;
    *(float4*)(&Bbuf[0][brow1 * B_STRIDE + bcol1]) = bf1;
    if (tid < 64) *(float4*)(&Abuf[0][arow * A_STRIDE + acol]) = af;
    __syncthreads();

    v8f c = {0.f, 0.f, 0.f, 0.f, 0.f, 0.f, 0.f, 0.f};
    int p = 0;

    for (int ch = 0; ch < CC / KCHUNK; ++ch) {            // 15 chunks
        // issue global loads for next chunk (overlaps with WMMAs below)
        const int kc2 = (ch + 1) * KCHUNK;
        if (ch < CC / KCHUNK - 1) {
            bf0 = *(const float4*)(W + (size_t)(kc2 + brow0) * HD + nb0 + bcol0);
            bf1 = *(const float4*)(W + (size_t)(kc2 + brow1) * HD + nb0 + bcol1);
            if (tid < 64) af = *(const float4*)(feat + (size_t)(m0 + arow) * CIN + kc2 + acol);
        }

        // 4 WMMAs from LDS buffer p
        const float* __restrict__ Ab = &Abuf[p][0];
        const float* __restrict__ Bb = &Bbuf[p][0];
        #pragma unroll
        for (int kk = 0; kk < KCHUNK; kk += 4) {
            const int k = kk + 2 * half;
            v2f a = *(const v2f*)(Ab + r * A_STRIDE + k);      // ds_load_b64, conflict-free
            v2f b;
            b.x = Bb[k * B_STRIDE + wave * 16 + r];            // ds_load_b32
            b.y = Bb[(k + 1) * B_STRIDE + wave * 16 + r];
            c = __builtin_amdgcn_wmma_f32_16x16x4_f32(false, a, false, b,
                                                      (short)0, c, false, false);
        }

        // drain next chunk into the other buffer
        if (ch < CC / KCHUNK - 1) {
            *(float4*)(&Bbuf[1 - p][brow0 * B_STRIDE + bcol0]) = bf0;
            *(float4*)(&Bbuf[1 - p][brow1 * B_STRIDE + bcol1]) = bf1;
            if (tid < 64) *(float4*)(&Abuf[1 - p][arow * A_STRIDE + acol]) = af;
            __syncthreads();
            p ^= 1;
        }
    }

    // C/D layout: VGPR rr -> M = rr + 8*half, N = lane&15
    #pragma unroll
    for (int rr = 0; rr < 8; ++rr) {
        ft[(size_t)(m0 + rr + 8 * half) * HD + n0 + r] = c[rr];
    }
}

// ---------- per-node scores: ssrc/sdst = <ft, attn> + <pos, pos_attn> ----------
__global__ void scores_kernel(const float* __restrict__ ft, const float* __restrict__ feat,
                              const float* __restrict__ attn_src, const float* __restrict__ attn_dst,
                              const float* __restrict__ pos_attn_src, const float* __restrict__ pos_attn_dst,
                              float* __restrict__ ssrc, float* __restrict__ sdst) {
    int i = blockIdx.x * blockDim.x + threadIdx.x;
    if (i >= NN * HH) return;
    const int n = i >> 2;
    const int h = i & 3;
    const float* __restrict__ fr = ft + (size_t)n * HD + h * DD;
    const float* __restrict__ as = attn_src + h * DD;
    const float* __restrict__ ad = attn_dst + h * DD;
    float ss = 0.f, sd = 0.f;
    #pragma unroll 8
    for (int d = 0; d < DD; ++d) {
        float v = fr[d];
        ss += v * as[d];
        sd += v * ad[d];
    }
    const float* __restrict__ pr = feat + (size_t)n * CIN + CC;   // last 16 = pos
    const float* __restrict__ ps = pos_attn_src + h * PP;
    const float* __restrict__ pd = pos_attn_dst + h * PP;
    #pragma unroll
    for (int pq = 0; pq < PP; ++pq) {
        float v = pr[pq];
        ss += v * ps[pq];
        sd += v * pd[pq];
    }
    ssrc[i] = ss;
    sdst[i] = sd;
}

// ---------- edge pass 1: leaky-relu logits + segment max ----------
__global__ void edge_logits_kernel(const int* __restrict__ src, const int* __restrict__ dst,
                                   const float* __restrict__ ssrc, const float* __restrict__ sdst,
                                   float* __restrict__ elog, unsigned* __restrict__ emaxk) {
    int i = blockIdx.x * blockDim.x + threadIdx.x;
    if (i >= EE * HH) return;
    const int e = i >> 2;
    const int h = i & 3;
    const int s = src[e];
    const int d = dst[e];
    float v = ssrc[s * HH + h] + sdst[d * HH + h];
    v = (v > 0.f) ? v : 0.2f * v;                 // leaky_relu, slope 0.2
    elog[i] = v;
    atomicMax(&emaxk[d * HH + h], enc_f32(v));
}

// ---------- edge pass 2: exp(e - max) + segment denom ----------
__global__ void edge_exp_kernel(const int* __restrict__ dst,
                                float* __restrict__ elog,
                                const unsigned* __restrict__ emaxk,
                                float* __restrict__ denom) {
    int i = blockIdx.x * blockDim.x + threadIdx.x;
    if (i >= EE * HH) return;
    const int e = i >> 2;
    const int h = i & 3;
    const int d = dst[e];
    float m = dec_f32(emaxk[d * HH + h]);
    float ex = expf(elog[i] - m);
    elog[i] = ex;
    atomicAdd(&denom[d * HH + h], ex);
}

// ---------- edge pass 3: scatter ft[src] * a into out[dst] ----------
// warp per edge; lane handles 8 consecutive cols (all within one head)
__global__ void __launch_bounds__(256) edge_scatter_kernel(const int* __restrict__ src,
                                                           const int* __restrict__ dst,
                                                           const float* __restrict__ elog,
                                                           const float* __restrict__ denom,
                                                           const float* __restrict__ ft,
                                                           float* __restrict__ out) {
    const int wid = blockIdx.x * 8 + (threadIdx.x >> 5);
    const int lane = threadIdx.x & 31;
    if (wid >= EE) return;
    const int s = src[wid];
    const int d = dst[wid];
    const int h = lane >> 3;                       // lane*8 in [h*64, h*64+63]
    const float a = elog[wid * HH + h] / denom[d * HH + h];

    const float4* __restrict__ fs = (const float4*)(ft + (size_t)s * HD + lane * 8);
    float4 v0 = fs[0];
    float4 v1 = fs[1];
    float* __restrict__ po = out + (size_t)d * HD + lane * 8;
    atomicAdd(po + 0, v0.x * a);
    atomicAdd(po + 1, v0.y * a);
    atomicAdd(po + 2, v0.z * a);
    atomicAdd(po + 3, v0.w * a);
    atomicAdd(po + 4, v1.x * a);
    atomicAdd(po + 5, v1.y * a);
    atomicAdd(po + 6, v1.z * a);
    atomicAdd(po + 7, v1.w * a);
}

extern "C" void kernel_launch(void* const* d_in, const int* in_sizes, int n_in,
                              void* d_out, int out_size, void* d_ws, size_t ws_size,
                              hipStream_t stream) {
    const float* feat         = (const float*)d_in[0];
    const int*   src          = (const int*)  d_in[1];
    const int*   dst          = (const int*)  d_in[2];
    const float* W            = (const float*)d_in[3];
    const float* attn_src     = (const float*)d_in[4];
    const float* attn_dst     = (const float*)d_in[5];
    const float* pos_attn_src = (const float*)d_in[6];
    const float* pos_attn_dst = (const float*)d_in[7];
    float* out = (float*)d_out;

    // workspace layout (~67.2 MB)
    float*    ft    = (float*)d_ws;                        // N * 256
    float*    ssrc  = ft + (size_t)NN * HD;                // N * 4
    float*    sdst  = ssrc + (size_t)NN * HH;              // N * 4
    unsigned* emaxk = (unsigned*)(sdst + (size_t)NN * HH); // N * 4
    float*    denom = (float*)(emaxk + (size_t)NN * HH);   // N * 4
    float*    elog  = denom + (size_t)NN * HH;             // E * 4

    init_out_kernel<<<(NN * CIN / 4 + 255) / 256, 256, 0, stream>>>((const float4*)feat, (float4*)out);
    init_nh_kernel<<<(NN * HH + 255) / 256, 256, 0, stream>>>(emaxk, denom);

    gemm_wmma_kernel<<<(NN / 16) * 2, 256, 0, stream>>>(feat, W, ft);  // 6250 blocks

    scores_kernel<<<(NN * HH + 255) / 256, 256, 0, stream>>>(ft, feat, attn_src, attn_dst,
                                                             pos_attn_src, pos_attn_dst, ssrc, sdst);

    edge_logits_kernel<<<(EE * HH + 255) / 256, 256, 0, stream>>>(src, dst, ssrc, sdst, elog, emaxk);
    edge_exp_kernel<<<(EE * HH + 255) / 256, 256, 0, stream>>>(dst, elog, emaxk, denom);
    edge_scatter_kernel<<<EE / 8, 256, 0, stream>>>(src, dst, elog, denom, ft, out);
}